// Condnet_model_33045478375694
// MI455X (gfx1250) — compile-verified
//
#include <hip/hip_runtime.h>
#include <math.h>

typedef __attribute__((ext_vector_type(2))) float v2f;
typedef __attribute__((ext_vector_type(8))) float v8f;

#define WMMA_F32(A, B, C) \
  __builtin_amdgcn_wmma_f32_16x16x4_f32(false, (A), false, (B), (short)0, (C), false, false)

// ---------------------------------------------------------------------------
// K1: one adj pass -> deg[b,n] = max(rowsum,1), alpha[b,n] = (adj@h)/deg
// One wave per (b,n) row; 8 rows per 256-thread block; float4 streaming loads.
// ---------------------------------------------------------------------------
__global__ __launch_bounds__(256) void k1_deg_alpha(
    const float* __restrict__ adj, const float* __restrict__ h,
    float* __restrict__ deg, float* __restrict__ alpha, int N) {
  int row = blockIdx.x * 8 + (threadIdx.x >> 5);   // row in [0, B*N)
  int lane = threadIdx.x & 31;
  int b = row / N;
  const float4* a = (const float4*)(adj + (size_t)row * N);
  const float4* hb = (const float4*)(h + (size_t)b * N);
  int n4 = N >> 2;
  float sd = 0.f, sa = 0.f;
  for (int k = lane; k < n4; k += 32) {
    float4 av = a[k];
    float4 hv = hb[k];
    sd += (av.x + av.y) + (av.z + av.w);
    sa += av.x * hv.x + av.y * hv.y + av.z * hv.z + av.w * hv.w;
  }
#pragma unroll
  for (int off = 16; off; off >>= 1) {
    sd += __shfl_xor(sd, off);
    sa += __shfl_xor(sa, off);
  }
  if (lane == 0) {
    float dg = sd < 1.f ? 1.f : sd;
    deg[row] = dg;
    alpha[row] = sa / dg;
  }
}

// ---------------------------------------------------------------------------
// K2: per workgroup: one batch b, one 16-row tile m0.
//   AGG = adj[b, m0:m0+16, :] @ z1[b]            (fp32 WMMA, K-loop over N)
//   z1 recomputed on the fly: z1[k,c] = relu(alpha[k]*Wr1[c] + h[k]*Wt1[c] + b1[c])
//   z2 = relu(AGG/deg @ Wr2 + z1[m-rows] @ Wt2 + b2)   (fp32 WMMA, K=64)
//   s2[m] = z2[m,:]·Wr3 ; t2[m] = z2[m,:]·Wt3
// 128 threads = 4 waves; wave w owns output columns [16w,16w+16).
// ---------------------------------------------------------------------------
__global__ __launch_bounds__(128) void k2_sage(
    const float* __restrict__ adj, const float* __restrict__ h,
    const float* __restrict__ alpha, const float* __restrict__ deg,
    const float* __restrict__ Wr1, const float* __restrict__ Wt1,
    const float* __restrict__ br1, const float* __restrict__ bt1,
    const float* __restrict__ Wr2, const float* __restrict__ Wt2,
    const float* __restrict__ br2, const float* __restrict__ bt2,
    const float* __restrict__ Wr3, const float* __restrict__ Wt3,
    float* __restrict__ s2, float* __restrict__ t2, int N) {
  __shared__ float sWr1[64], sWt1[64], sB1[64];
  __shared__ float sWr3[64], sWt3[64], sB2[64];
  __shared__ float sWr2[64][65], sWt2[64][65];
  __shared__ float adjt[16][17];
  __shared__ float z1t[16][65];
  __shared__ float aggt[16][65];
  __shared__ float z2t[16][65];
  __shared__ float sInvDeg[16];

  const int tid = threadIdx.x;
  const int wave = tid >> 5;
  const int lane = tid & 31;
  const int b = blockIdx.y;
  const int m0 = blockIdx.x * 16;
  const int n0 = wave * 16;
  const int hl = lane & 15;         // m (for A) or n-local (for B/C)
  const int hk = (lane >> 4) * 2;   // K sub-offset: 0 for lanes 0-15, 2 for 16-31

  for (int i = tid; i < 64; i += 128) {
    sWr1[i] = Wr1[i]; sWt1[i] = Wt1[i]; sB1[i] = br1[i] + bt1[i];
    sWr3[i] = Wr3[i]; sWt3[i] = Wt3[i]; sB2[i] = br2[i] + bt2[i];
  }
  for (int i = tid; i < 64 * 64; i += 128) {
    int r = i >> 6, c = i & 63;
    sWr2[r][c] = Wr2[i];
    sWt2[r][c] = Wt2[i];
  }
  __syncthreads();

  const float* adjb = adj + (size_t)b * N * N;
  const float* hb = h + (size_t)b * N;
  const float* alb = alpha + (size_t)b * N;
  const float* degb = deg + (size_t)b * N;

  v8f acc = {};  // 16(m) x 16(n0..) tile of adj@z1

  for (int k0 = 0; k0 < N; k0 += 16) {
    // stage adj tile [16 m][16 k]
    {
      int r = tid >> 3;
      int c = (tid & 7) * 2;
      const float* src = adjb + (size_t)(m0 + r) * N + (k0 + c);
      adjt[r][c] = src[0];
      adjt[r][c + 1] = src[1];
    }
    // stage z1 tile [16 k][64 c], recomputed from alpha/h (rank-2 + relu)
    {
      int kk = tid >> 3;
      float av = alb[k0 + kk];
      float hv = hb[k0 + kk];
      int nb = (tid & 7) * 8;
#pragma unroll
      for (int j = 0; j < 8; ++j) {
        int n = nb + j;
        float v = av * sWr1[n] + hv * sWt1[n] + sB1[n];
        z1t[kk][n] = v > 0.f ? v : 0.f;
      }
    }
    __syncthreads();
#pragma unroll
    for (int g = 0; g < 4; ++g) {
      int kb = g * 4;
      v2f A, Bv;
      A.x = adjt[hl][kb + hk];        // A: lanes0-15 K={0,1}, lanes16-31 K={2,3}
      A.y = adjt[hl][kb + hk + 1];
      Bv.x = z1t[kb + hk][n0 + hl];   // B mirrored layout
      Bv.y = z1t[kb + hk + 1][n0 + hl];
      acc = WMMA_F32(A, Bv, acc);
    }
    __syncthreads();
  }

  // reciprocal degree once per tile row (avoids 8 IEEE divides per lane)
  if (tid < 16) sInvDeg[tid] = 1.f / degb[m0 + tid];
  __syncthreads();
#pragma unroll
  for (int r = 0; r < 8; ++r) {
    int m = r + ((lane >> 4) << 3);
    aggt[m][n0 + hl] = acc[r] * sInvDeg[m];
  }
  __syncthreads();
  // recompute z1 rows m0..m0+15 (root term)
  {
    int kk = tid >> 3;
    float av = alb[m0 + kk];
    float hv = hb[m0 + kk];
    int nb = (tid & 7) * 8;
#pragma unroll
    for (int j = 0; j < 8; ++j) {
      int n = nb + j;
      float v = av * sWr1[n] + hv * sWt1[n] + sB1[n];
      z1t[kk][n] = v > 0.f ? v : 0.f;
    }
  }
  __syncthreads();

  // z2 = relu(AGG @ Wr2 + z1 @ Wt2 + b2), two K=64 WMMA chains
  v8f acc2 = {};
#pragma unroll
  for (int g = 0; g < 16; ++g) {
    int kb = g * 4;
    v2f A, Bv;
    A.x = aggt[hl][kb + hk];
    A.y = aggt[hl][kb + hk + 1];
    Bv.x = sWr2[kb + hk][n0 + hl];
    Bv.y = sWr2[kb + hk + 1][n0 + hl];
    acc2 = WMMA_F32(A, Bv, acc2);
    A.x = z1t[hl][kb + hk];
    A.y = z1t[hl][kb + hk + 1];
    Bv.x = sWt2[kb + hk][n0 + hl];
    Bv.y = sWt2[kb + hk + 1][n0 + hl];
    acc2 = WMMA_F32(A, Bv, acc2);
  }
#pragma unroll
  for (int r = 0; r < 8; ++r) {
    int m = r + ((lane >> 4) << 3);
    float v = acc2[r] + sB2[n0 + hl];
    z2t[m][n0 + hl] = v > 0.f ? v : 0.f;
  }
  __syncthreads();

  if (tid < 16) {
    float s = 0.f, t = 0.f;
#pragma unroll 8
    for (int c = 0; c < 64; ++c) {
      float z = z2t[tid][c];
      s += z * sWr3[c];
      t += z * sWt3[c];
    }
    s2[(size_t)b * N + m0 + tid] = s;
    t2[(size_t)b * N + m0 + tid] = t;
  }
}

// ---------------------------------------------------------------------------
// K3: one adj pass -> agg3 = (adj@s2)/deg; gate; hg = h * u
// ---------------------------------------------------------------------------
__global__ __launch_bounds__(256) void k3_gate(
    const float* __restrict__ adj, const float* __restrict__ s2,
    const float* __restrict__ t2, const float* __restrict__ deg,
    const float* __restrict__ h, const float* __restrict__ noise,
    const float* __restrict__ br3, const float* __restrict__ bt3,
    const float* __restrict__ fcW, const float* __restrict__ fcb,
    float* __restrict__ hg, int N) {
  int row = blockIdx.x * 8 + (threadIdx.x >> 5);
  int lane = threadIdx.x & 31;
  int b = row / N;
  const float4* a = (const float4*)(adj + (size_t)row * N);
  const float4* sb = (const float4*)(s2 + (size_t)b * N);
  int n4 = N >> 2;
  float s = 0.f;
  for (int k = lane; k < n4; k += 32) {
    float4 av = a[k];
    float4 sv = sb[k];
    s += av.x * sv.x + av.y * sv.y + av.z * sv.z + av.w * sv.w;
  }
#pragma unroll
  for (int off = 16; off; off >>= 1) s += __shfl_xor(s, off);
  if (lane == 0) {
    float z3 = s / deg[row] + t2[row] + br3[0] + bt3[0];
    z3 = z3 > 0.f ? z3 : 0.f;
    float logit = z3 * fcW[0] + fcb[0];
    float pr = 1.f / (1.f + expf(-logit));
    pr = fminf(fmaxf(pr, 1e-6f), 1.f - 1e-6f);
    pr = pr * 0.8f + 0.1f;                 // (MAXP-MINP)=0.8, +MINP
    float u = (noise[row] < pr) ? 1.f : 0.f;
    hg[row] = h[row] * u;
  }
}

// ---------------------------------------------------------------------------
// K4: dense MLP layer Y = relu(X @ W + b), fp32 WMMA; wave -> 16x16 tile
// ---------------------------------------------------------------------------
__global__ __launch_bounds__(128) void k4_mlp(
    const float* __restrict__ X, const float* __restrict__ W,
    const float* __restrict__ bias, float* __restrict__ Y, int K, int M) {
  int wave = threadIdx.x >> 5, lane = threadIdx.x & 31;
  int j0 = blockIdx.x * 64 + wave * 16;
  int b0 = blockIdx.y * 16;
  int hl = lane & 15, hk = (lane >> 4) * 2;
  v8f acc = {};
  const float* xr = X + (size_t)(b0 + hl) * K;
  for (int k = 0; k < K; k += 4) {
    v2f A, Bv;
    A.x = xr[k + hk];
    A.y = xr[k + hk + 1];
    Bv.x = W[(size_t)(k + hk) * M + j0 + hl];
    Bv.y = W[(size_t)(k + hk + 1) * M + j0 + hl];
    acc = WMMA_F32(A, Bv, acc);
  }
  float bj = bias[j0 + hl];
#pragma unroll
  for (int r = 0; r < 8; ++r) {
    int m = r + ((lane >> 4) << 3);
    float v = acc[r] + bj;
    Y[(size_t)(b0 + m) * M + j0 + hl] = v > 0.f ? v : 0.f;
  }
}

// ---------------------------------------------------------------------------
// K5: last layer (256 -> 10) + softmax over classes; one wave per batch row
// ---------------------------------------------------------------------------
__global__ __launch_bounds__(32) void k5_final(
    const float* __restrict__ X, const float* __restrict__ W,
    const float* __restrict__ bias, float* __restrict__ out) {
  int b = blockIdx.x;
  int lane = threadIdx.x;
  float v = -1e30f;
  if (lane < 10) {
    float s = bias[lane];
    const float* xr = X + (size_t)b * 256;
    for (int k = 0; k < 256; ++k) s += xr[k] * W[k * 10 + lane];
    v = s;
  }
  float m = v;
#pragma unroll
  for (int off = 16; off; off >>= 1) m = fmaxf(m, __shfl_xor(m, off));
  float e = (lane < 10) ? expf(v - m) : 0.f;
  float sum = e;
#pragma unroll
  for (int off = 16; off; off >>= 1) sum += __shfl_xor(sum, off);
  if (lane < 10) out[b * 10 + lane] = e / sum;
}

// ---------------------------------------------------------------------------
extern "C" void kernel_launch(void* const* d_in, const int* in_sizes, int n_in,
                              void* d_out, int out_size, void* d_ws, size_t ws_size,
                              hipStream_t stream) {
  (void)in_sizes; (void)n_in; (void)out_size; (void)ws_size;
  const int B = 128;
  const int dims[4] = {784, 512, 256, 10};

  // setup_inputs() dict order:
  // 0:x 1:adj0 2:noise0 3:adj1 4:noise1 5:adj2 6:noise2
  // 7:W0 8:b0 9:W1 10:b1 11:W2 12:b2
  // 13+14*li: Wr1 Wt1 br1 bt1 Wr2 Wt2 br2 bt2 Wr3 Wt3 br3 bt3 fcW fcb
  const float* x = (const float*)d_in[0];
  const float* adjL[3] = {(const float*)d_in[1], (const float*)d_in[3], (const float*)d_in[5]};
  const float* noiseL[3] = {(const float*)d_in[2], (const float*)d_in[4], (const float*)d_in[6]};
  const float* Wm[3] = {(const float*)d_in[7], (const float*)d_in[9], (const float*)d_in[11]};
  const float* bm[3] = {(const float*)d_in[8], (const float*)d_in[10], (const float*)d_in[12]};

  float* w = (float*)d_ws;
  const size_t NB = (size_t)B * 784;
  float* deg = w;
  float* alpha = w + NB;
  float* s2 = w + 2 * NB;
  float* t2 = w + 3 * NB;
  float* hg = w + 4 * NB;
  float* h1 = w + 5 * NB;
  float* h2 = h1 + (size_t)B * 512;

  const float* h = x;
  for (int li = 0; li < 3; ++li) {
    int N = dims[li], M = dims[li + 1];
    int gi = 13 + 14 * li;
    const float* Wr1 = (const float*)d_in[gi + 0];
    const float* Wt1 = (const float*)d_in[gi + 1];
    const float* br1 = (const float*)d_in[gi + 2];
    const float* bt1 = (const float*)d_in[gi + 3];
    const float* Wr2 = (const float*)d_in[gi + 4];
    const float* Wt2 = (const float*)d_in[gi + 5];
    const float* br2 = (const float*)d_in[gi + 6];
    const float* bt2 = (const float*)d_in[gi + 7];
    const float* Wr3 = (const float*)d_in[gi + 8];
    const float* Wt3 = (const float*)d_in[gi + 9];
    const float* br3 = (const float*)d_in[gi + 10];
    const float* bt3 = (const float*)d_in[gi + 11];
    const float* fcW = (const float*)d_in[gi + 12];
    const float* fcb = (const float*)d_in[gi + 13];

    k1_deg_alpha<<<(B * N) / 8, 256, 0, stream>>>(adjL[li], h, deg, alpha, N);
    k2_sage<<<dim3(N / 16, B), 128, 0, stream>>>(adjL[li], h, alpha, deg,
                                                 Wr1, Wt1, br1, bt1,
                                                 Wr2, Wt2, br2, bt2,
                                                 Wr3, Wt3, s2, t2, N);
    k3_gate<<<(B * N) / 8, 256, 0, stream>>>(adjL[li], s2, t2, deg, h, noiseL[li],
                                             br3, bt3, fcW, fcb, hg, N);
    if (li < 2) {
      float* hn = (li == 0) ? h1 : h2;
      k4_mlp<<<dim3(M / 64, B / 16), 128, 0, stream>>>(hg, Wm[li], bm[li], hn, N, M);
      h = hn;
    } else {
      k5_final<<<B, 32, 0, stream>>>(hg, Wm[2], bm[2], (float*)d_out);
    }
  }
}